// CategoricalLoss_70866960384578
// MI455X (gfx1250) — compile-verified
//
#include <hip/hip_runtime.h>
#include <stdint.h>

#define ATOMS 51
#define ROWS 128            // rows per block == threads per block (4 waves of 32)
#define VMIN (-10.0f)
#define VMAXF (10.0f)
#define DELTAF (0.4f)

typedef float v2f  __attribute__((ext_vector_type(2)));
typedef float v8f  __attribute__((ext_vector_type(8)));
typedef unsigned int u32x4 __attribute__((ext_vector_type(4)));
typedef int   i32x4 __attribute__((ext_vector_type(4)));
typedef int   i32x8 __attribute__((ext_vector_type(8)));

typedef __attribute__((address_space(3))) char lds_char;

// Byte offset of a (generic) pointer into LDS, as the TDM descriptor wants it.
__device__ __forceinline__ uint32_t lds_addr_of(void* p) {
  return (uint32_t)(unsigned long long)(lds_char*)p;
}

// Issue one TDM 2D tile load: rows x ATOMS fp32, row pitch == ATOMS (dense),
// global tile start -> LDS at the byte offset in lds_byte.  Descriptor packing per CDNA5 ISA §8.
// This toolchain exposes the 6-arg builtin:
//   (uint32x4 g0, int32x8 g1, int32x4 g2, int32x4 g3, int32x8 extra, i32 cpol)
__device__ __forceinline__ void tdm_load_2d(const void* gsrc, uint32_t lds_byte,
                                            uint32_t rows) {
  unsigned long long ga = (unsigned long long)gsrc;
  u32x4 g0;
  g0[0] = 1u;                                            // count=1 (valid D#)
  g0[1] = lds_byte;                                      // lds_addr [63:32]
  g0[2] = (uint32_t)(ga & 0xFFFFFFFFu);                  // global_addr [95:64]
  g0[3] = (uint32_t)((ga >> 32) & 0x01FFFFFFu)           // global_addr [120:96]
        | (2u << 30);                                    // type=2 ("image") [127:126]
  i32x8 g1;
  g1[0] = (int)(2u << 16);          // data_size=2 (4 bytes); wg_mask=0
  g1[1] = (int)(ATOMS << 16);       // tensor_dim0[15:0] at bits 63:48
  g1[2] = (int)(rows  << 16);       // tensor_dim1[15:0] at bits 95:80
  g1[3] = (int)(ATOMS << 16);       // tile_dim0 at bits 127:112
  g1[4] = (int)rows;                // tile_dim1 at bits 143:128
  g1[5] = ATOMS;                    // tensor_dim0_stride[31:0]
  g1[6] = 0;                        // stride hi / dim1_stride lo (unused, 2D)
  g1[7] = 0;
  i32x4 z4 = {0, 0, 0, 0};          // groups 2/3 unused for 2D tiles
  i32x8 z8 = {0, 0, 0, 0, 0, 0, 0, 0};
  __builtin_amdgcn_tensor_load_to_lds(g0, g1, z4, z4, z8, 0);
}

__global__ __launch_bounds__(ROWS)
void c51_loss_partial(const float* __restrict__ anchor,
                      const float* __restrict__ feature,
                      const float* __restrict__ skew_p,
                      const int*   __restrict__ direction,
                      const float* __restrict__ weight,
                      float* __restrict__ partials) {
  __shared__ __align__(16) float  ldsA[ROWS * ATOMS];   // anchor tile
  __shared__ __align__(16) float  ldsF[ROWS * ATOMS];   // feature tile -> logs
  __shared__ __align__(16) float4 tbl[2][ATOMS];        // {as_int(l), wl, wu, -}
  __shared__ float red[ROWS];

  const int tid = threadIdx.x;
  const int r0  = blockIdx.x * ROWS;

  // Wave 0 kicks off both TDM tile DMAs immediately.
  if (tid < 32) {
    tdm_load_2d(anchor  + (size_t)r0 * ATOMS, lds_addr_of(ldsA), ROWS);
    tdm_load_2d(feature + (size_t)r0 * ATOMS, lds_addr_of(ldsF), ROWS);
  }

  // Projection table (depends only on direction, 2 cases x 51 atoms), built
  // while the DMA is in flight.  Mirrors the reference op sequence exactly,
  // including the divide by DELTA.  Note u == l+1 always after adjustment.
  const float s = *skew_p;
  if (tid < 2 * ATOMS) {
    const int d = (tid >= ATOMS);
    const int j = tid - d * ATOMS;
    const float skew = d ? -s : s;
    float sup = fmaf((float)j, DELTAF, VMIN);
    float Tz  = fminf(fmaxf(skew + sup, VMIN), VMAXF);
    float bb  = (Tz - VMIN) / DELTAF;
    int l = (int)floorf(bb);
    int u = (int)ceilf(bb);
    if (l == u) {               // eq case: l-1 if u>0, else u+1 (b==0)
      if (u > 0) --l; else ++u;
    }
    float4 e;
    e.x = __int_as_float(l);
    e.y = (float)u - bb;        // wl
    e.z = bb - (float)l;        // wu
    e.w = 0.0f;
    tbl[d][j] = e;
  }

  if (tid < 32) __builtin_amdgcn_s_wait_tensorcnt(0);
  __syncthreads();

  // Pass 1: log(feature + 1e-16) in place, elementwise over the flat tile
  // with float4 LDS accesses (6528 dwords == 1632 b128 ops, conflict-free).
  {
    float4* f4 = (float4*)ldsF;
    for (int i = tid; i < (ROWS * ATOMS) / 4; i += ROWS) {
      float4 v = f4[i];
      v.x = __logf(v.x + 1e-16f);
      v.y = __logf(v.y + 1e-16f);
      v.z = __logf(v.z + 1e-16f);
      v.w = __logf(v.w + 1e-16f);
      f4[i] = v;
    }
  }
  __syncthreads();

  // Pass 2: gathered projection sum, one row per thread.
  const int   g_row  = r0 + tid;
  const int   dirIdx = (direction[g_row] == 1) ? 1 : 0;
  const float4* trow = tbl[dirIdx];
  const float*  fr   = &ldsF[tid * ATOMS];
  const float*  ar   = &ldsA[tid * ATOMS];

  float acc = 0.0f;
  for (int j = 0; j < ATOMS; ++j) {
    float4 e = trow[j];
    int l = __float_as_int(e.x);
    acc = fmaf(ar[j], fmaf(e.y, fr[l], e.z * fr[l + 1]), acc);
  }
  red[tid] = acc * weight[g_row];
  __syncthreads();

  // Block reduction of 128 values using V_WMMA_F32_16X16X4_F32 with a ones
  // B matrix: D[m][n] accumulates per-m sums; two chained WMMAs cover 128.
  if (tid < 32) {
    v8f c = {0.f, 0.f, 0.f, 0.f, 0.f, 0.f, 0.f, 0.f};
    v2f ones; ones[0] = 1.0f; ones[1] = 1.0f;
#pragma unroll
    for (int ch = 0; ch < ROWS / 64; ++ch) {
      v2f a;
      a[0] = red[ch * 64 + tid];
      a[1] = red[ch * 64 + 32 + tid];
      c = __builtin_amdgcn_wmma_f32_16x16x4_f32(
              /*neg_a=*/false, a, /*neg_b=*/false, ones,
              /*c_mod=*/(short)0, c, /*reuse_a=*/false, /*reuse_b=*/false);
    }
    // Lane n (0..15) holds D[0..7][n] in c[0..7]; lane 16+n holds D[8..15][n].
    float ssum = c[0] + c[1] + c[2] + c[3] + c[4] + c[5] + c[6] + c[7];
    float tot  = ssum + __shfl(ssum, 16, 32);
    if (tid == 0) partials[blockIdx.x] = tot;
  }
}

__global__ __launch_bounds__(256)
void c51_finalize(const float* __restrict__ partials, int n,
                  float* __restrict__ out, float negInvB) {
  __shared__ float sm[256];
  float a = 0.0f;
  for (int i = threadIdx.x; i < n; i += 256) a += partials[i];
  sm[threadIdx.x] = a;
  __syncthreads();
  for (int off = 128; off > 0; off >>= 1) {
    if ((int)threadIdx.x < off) sm[threadIdx.x] += sm[threadIdx.x + off];
    __syncthreads();
  }
  if (threadIdx.x == 0) out[0] = sm[0] * negInvB;
}

extern "C" void kernel_launch(void* const* d_in, const int* in_sizes, int n_in,
                              void* d_out, int out_size, void* d_ws, size_t ws_size,
                              hipStream_t stream) {
  const float* anchor  = (const float*)d_in[0];
  const float* feature = (const float*)d_in[1];
  const float* skew    = (const float*)d_in[2];
  const int*   dir     = (const int*)d_in[3];
  const float* weight  = (const float*)d_in[4];
  float* out      = (float*)d_out;
  float* partials = (float*)d_ws;

  const int B = in_sizes[3];          // direction has B elements
  const int nblocks = B / ROWS;       // B = 524288 -> 4096 blocks

  c51_loss_partial<<<nblocks, ROWS, 0, stream>>>(anchor, feature, skew, dir,
                                                 weight, partials);
  c51_finalize<<<1, 256, 0, stream>>>(partials, nblocks, out,
                                      -1.0f / (float)B);
}